// AdaptiveLaplacian_88278757802290
// MI455X (gfx1250) — compile-verified
//
#include <hip/hip_runtime.h>
#include <math.h>

typedef __attribute__((ext_vector_type(2))) float v2f;
typedef __attribute__((ext_vector_type(4))) float v4f;
typedef __attribute__((ext_vector_type(8))) float v8f;

#define NPTS 131072
#define CCH  128
#define NSMP 16
#define GEMM_BLOCKS (NPTS / 128)   // 1024
static constexpr float BN_EPS = 1e-5f;

// ---------------------------------------------------------------------------
// Y[N,128] = X[N,128] @ W[128,128] + bias  (fp32 WMMA 16x16x4)
// If `partials` != nullptr also emit per-block BN partial sums:
//   partials[blk*256 + c]       = sum over block rows of Y[:,c]
//   partials[blk*256 + 128 + c] = sum of squares
// Block: 256 threads (8 waves), each wave owns a 16-row x 128-col tile.
// ---------------------------------------------------------------------------
__global__ __launch_bounds__(256) void gemm128_wmma(
    const float* __restrict__ X, const float* __restrict__ W,
    const float* __restrict__ bias, float* __restrict__ Y,
    float* __restrict__ partials)
{
  // W transposed into LDS: sWt[j*130 + k] = W[k*128 + j]
  // stride 130 (even) -> bank-conflict-free for both staging and B-frag b64 reads
  __shared__ float sWt[128 * 130];

  const int tid = threadIdx.x;
  for (int i = tid; i < 128 * 128; i += 256) {
    const int k = i >> 7, j = i & 127;
    sWt[j * 130 + k] = W[i];
  }
  __syncthreads();

  const int wave = tid >> 5;
  const int lane = tid & 31;
  const int m  = lane & 15;   // row within 16-row tile (A), col within 16-col tile (B/C)
  const int hi = lane >> 4;   // K-half selector per 32-bit WMMA layouts
  const int rowBase = blockIdx.x * 128 + wave * 16;
  const float* xrow = X + (size_t)(rowBase + m) * CCH + 2 * hi;

  v8f acc[8];
#pragma unroll
  for (int jt = 0; jt < 8; ++jt) {
    const float bv = bias[jt * 16 + m];          // C/D element (M,N=m) -> bias[N]
    acc[jt] = (v8f){bv, bv, bv, bv, bv, bv, bv, bv};
  }

#pragma unroll 4
  for (int k = 0; k < 128; k += 4) {
    // A frag (16x4 f32): lane<16 holds K=k,k+1 of row m; lane>=16 holds K=k+2,k+3
    const v2f A = *(const v2f*)(xrow + k);
#pragma unroll
    for (int jt = 0; jt < 8; ++jt) {
      // B frag (4x16 f32): v0={K,K+2}, v1={K+1,K+3}, N=m -> contiguous pair in sWt
      const v2f B = *(const v2f*)(&sWt[(jt * 16 + m) * 130 + k + 2 * hi]);
      acc[jt] = __builtin_amdgcn_wmma_f32_16x16x4_f32(
          /*neg_a=*/false, A, /*neg_b=*/false, B,
          /*c_mod=*/(short)0, acc[jt], /*reuse_a=*/false, /*reuse_b=*/false);
    }
  }

  // C/D layout: VGPR r -> M = r + 8*hi, N = m
#pragma unroll
  for (int jt = 0; jt < 8; ++jt)
#pragma unroll
    for (int r = 0; r < 8; ++r)
      Y[(size_t)(rowBase + hi * 8 + r) * CCH + jt * 16 + m] = acc[jt][r];

  if (partials) {
    __syncthreads();                 // all waves done reading sWt -> reuse as scratch
    float* sc = sWt;                 // [16][128] row-sums
    float* sq = sWt + 16 * 128;      // [16][128] row-sumsq
#pragma unroll
    for (int jt = 0; jt < 8; ++jt) {
      float s = 0.f, q = 0.f;
#pragma unroll
      for (int r = 0; r < 8; ++r) { const float v = acc[jt][r]; s += v; q += v * v; }
      sc[(wave * 2 + hi) * 128 + jt * 16 + m] = s;
      sq[(wave * 2 + hi) * 128 + jt * 16 + m] = q;
    }
    __syncthreads();
    if (tid < 128) {                 // fixed-order reduction -> deterministic
      float s = 0.f, q = 0.f;
      for (int w = 0; w < 16; ++w) { s += sc[w * 128 + tid]; q += sq[w * 128 + tid]; }
      partials[blockIdx.x * 256 + tid]       = s;
      partials[blockIdx.x * 256 + 128 + tid] = q;
    }
  }
}

// ---------------------------------------------------------------------------
// lap[n,:] = mean_s ReLU(u_pre[idx[n,s],:] - u_pre[n,:])
// One wave per point (sequenced 16 points/wave); lane owns 4 channels (b128).
// Gathers hit L2 (u_pre = 64MB < 192MB L2).
// ---------------------------------------------------------------------------
__global__ __launch_bounds__(256) void laplacian_kernel(
    const float* __restrict__ upre, const int* __restrict__ idx,
    float* __restrict__ lap)
{
  const int wave = threadIdx.x >> 5;
  const int lane = threadIdx.x & 31;
  const int c4 = lane * 4;
  const int n0 = blockIdx.x * 128 + wave * 16;
  for (int pp = 0; pp < 16; ++pp) {
    const int n = n0 + pp;
    const v4f ctr = *(const v4f*)(upre + (size_t)n * CCH + c4);
    v4f acc = (v4f){0.f, 0.f, 0.f, 0.f};
    const int* ip = idx + (size_t)n * NSMP;
#pragma unroll
    for (int s = 0; s < NSMP; ++s) {
      const int j = ip[s];
      const v4f nb = *(const v4f*)(upre + (size_t)j * CCH + c4);
      const v4f d = nb - ctr;
      acc.x += fmaxf(d.x, 0.f);
      acc.y += fmaxf(d.y, 0.f);
      acc.z += fmaxf(d.z, 0.f);
      acc.w += fmaxf(d.w, 0.f);
    }
    *(v4f*)(lap + (size_t)n * CCH + c4) = acc * 0.0625f;
  }
}

// ---------------------------------------------------------------------------
// Reduce per-block partials -> per-channel scale/shift for fused BN.
// Single block, fixed order -> deterministic.
// ---------------------------------------------------------------------------
__global__ void bn_finalize(const float* __restrict__ partials, int nblk,
                            const float* __restrict__ gamma,
                            const float* __restrict__ beta,
                            float* __restrict__ scaleshift)
{
  const int c = threadIdx.x;  // 128 threads
  float s = 0.f, q = 0.f;
  for (int b = 0; b < nblk; ++b) {
    s += partials[b * 256 + c];
    q += partials[b * 256 + 128 + c];
  }
  const float inv_n = 1.0f / (float)NPTS;
  const float mean = s * inv_n;
  const float var  = q * inv_n - mean * mean;   // biased var, as BN uses
  const float sc   = gamma[c] * rsqrtf(var + BN_EPS);
  scaleshift[c]       = sc;
  scaleshift[128 + c] = beta[c] - mean * sc;
}

// u_tt = max(h*scale + shift, 0), in place over the h region of d_out
__global__ __launch_bounds__(256) void bn_relu_inplace(
    float* __restrict__ h, const float* __restrict__ scaleshift)
{
  const int gid = blockIdx.x * blockDim.x + threadIdx.x;  // over N*C/4
  const int i4 = gid * 4;
  const int c4 = i4 & (CCH - 1);
  const v4f sc = *(const v4f*)(scaleshift + c4);
  const v4f sh = *(const v4f*)(scaleshift + 128 + c4);
  v4f v = *(const v4f*)(h + i4);
  v = v * sc + sh;
  v.x = fmaxf(v.x, 0.f);
  v.y = fmaxf(v.y, 0.f);
  v.z = fmaxf(v.z, 0.f);
  v.w = fmaxf(v.w, 0.f);
  *(v4f*)(h + i4) = v;
}

// Pass-through outputs: p (N*3 floats) and o (1 int, bit-copied)
__global__ void copy_p_o(const float* __restrict__ p, const int* __restrict__ o,
                         float* __restrict__ out)
{
  const int gid = blockIdx.x * blockDim.x + threadIdx.x;
  const int n4 = (NPTS * 3) / 4;
  if (gid < n4) {
    ((v4f*)out)[gid] = ((const v4f*)p)[gid];
  } else if (gid == n4) {
    ((int*)out)[NPTS * 3 + NPTS * CCH] = o[0];
  }
}

extern "C" void kernel_launch(void* const* d_in, const int* in_sizes, int n_in,
                              void* d_out, int out_size, void* d_ws, size_t ws_size,
                              hipStream_t stream)
{
  const float* p     = (const float*)d_in[0];
  const float* u     = (const float*)d_in[1];
  const int*   o     = (const int*)  d_in[2];
  const int*   idx   = (const int*)  d_in[3];
  const float* W1    = (const float*)d_in[4];
  const float* b1    = (const float*)d_in[5];
  const float* W2    = (const float*)d_in[6];
  const float* b2    = (const float*)d_in[7];
  const float* gamma = (const float*)d_in[8];
  const float* beta  = (const float*)d_in[9];

  float* out = (float*)d_out;
  float* ws  = (float*)d_ws;

  float* upre       = ws;                                    // N*C
  float* lap        = ws + (size_t)NPTS * CCH;               // N*C
  float* partials   = ws + (size_t)2 * NPTS * CCH;           // GEMM_BLOCKS*256
  float* scaleshift = partials + (size_t)GEMM_BLOCKS * 256;  // 256
  float* h          = out + (size_t)NPTS * 3;                // u_tt slot of d_out

  gemm128_wmma<<<GEMM_BLOCKS, 256, 0, stream>>>(u, W1, b1, upre, nullptr);
  laplacian_kernel<<<NPTS / 128, 256, 0, stream>>>(upre, idx, lap);
  gemm128_wmma<<<GEMM_BLOCKS, 256, 0, stream>>>(lap, W2, b2, h, partials);
  bn_finalize<<<1, 128, 0, stream>>>(partials, GEMM_BLOCKS, gamma, beta, scaleshift);
  bn_relu_inplace<<<(NPTS * CCH / 4) / 256, 256, 0, stream>>>(h, scaleshift);
  copy_p_o<<<((NPTS * 3) / 4 + 1 + 255) / 256, 256, 0, stream>>>(p, o, out);
}